// so3_tensor_product_23837068492876
// MI455X (gfx1250) — compile-verified
//
#include <hip/hip_runtime.h>

// ============================================================================
// SO(3) tensor product on gfx1250 via V_WMMA_F32_16X16X4_F32.
//
// Per (b,i) element: z_k = x1[m]*x2[n] (81 products), out = W^T z where W is
// block-diagonal (9 square blocks). Blocks are packed into 5 "super-GEMMs":
//   SG0: (1,1)@0, (0,1)@9   -> K=N=12, KC=3, NT=1
//   SG1: (1,2)@0, (0,0)@15  -> K=N=16, KC=4, NT=1
//   SG2: (2,1)@0            -> K=N=15, KC=4, NT=1
//   SG3: (2,2)@0, (1,0)@25  -> K=N=28, KC=7, NT=2
//   SG4: (0,2)@0, (2,0)@5   -> K=N=10, KC=3, NT=1
// => 28 wmma_f32_16x16x4 per 16-row tile, 65536 tiles (1 wave each).
//
// WMMA f32 layouts assumed (CDNA5 ISA 7.12.2):
//   A 16x4 : lanes0-15 row M, v0=K0 v1=K1 ; lanes16-31 row M, v0=K2 v1=K3
//   B 4x16 : lanes0-15 col N, v0=K0 v1=K1 ; lanes16-31 col N, v0=K2 v1=K3
//   C 16x16: vgpr j: lanes0-15 (row j, col lane); lanes16-31 (row j+8, col lane-16)
// B values are built by tp_setup into d_ws in exactly this lane layout.
//
// Stores: C tiles are staged in per-wave LDS stage[16][84] (81 canonical
// columns = concat of SG valid columns), then each of the 19 (l1,l2,l3) path
// segments is a CONTIGUOUS run of 16*(2l3+1) floats in global memory
// (b fixed per tile, i0 16-aligned) -> fully coalesced 128B stores.
// ============================================================================

typedef __attribute__((ext_vector_type(2))) float v2f;
typedef __attribute__((ext_vector_type(8))) float v8f;

#define DEVFN __device__ __forceinline__

// ---------------- static-for (guarantees compile-time indices) --------------
template<int I> struct IC { static constexpr int v = I; };
template<int I, int N, typename F>
DEVFN void sfor(F&& f) {
  if constexpr (I < N) { f(IC<I>{}); if constexpr (I + 1 < N) sfor<I + 1, N>(f); }
}

// ---------------- compile-time tables for the main kernel -------------------
constexpr int SG_KC[5]    = {3, 4, 4, 7, 3};
constexpr int SG_NT[5]    = {1, 1, 1, 2, 1};
constexpr int SG_BOFF[5]  = {0, 192, 448, 704, 1600};   // floats into Bws
constexpr int SG_N[5]     = {12, 16, 15, 28, 10};       // valid columns
constexpr int SG_CBASE[5] = {0, 12, 28, 43, 71};        // stage column base

// Z index tables: k -> (i1,i2) into concat vectors [l0, l1[0..2], l2[0..4]].
// -1 = zero pad. Padded to 28 (max K).
constexpr int ZI1[5][28] = {
  {1,1,1,2,2,2,3,3,3,0,0,0,-1,-1,-1,-1,-1,-1,-1,-1,-1,-1,-1,-1,-1,-1,-1,-1},
  {1,1,1,1,1,2,2,2,2,2,3,3,3,3,3,0,-1,-1,-1,-1,-1,-1,-1,-1,-1,-1,-1,-1},
  {4,4,4,5,5,5,6,6,6,7,7,7,8,8,8,-1,-1,-1,-1,-1,-1,-1,-1,-1,-1,-1,-1,-1},
  {4,4,4,4,4,5,5,5,5,5,6,6,6,6,6,7,7,7,7,7,8,8,8,8,8,1,2,3},
  {0,0,0,0,0,4,5,6,7,8,-1,-1,-1,-1,-1,-1,-1,-1,-1,-1,-1,-1,-1,-1,-1,-1,-1,-1}
};
constexpr int ZI2[5][28] = {
  {1,2,3,1,2,3,1,2,3,1,2,3,-1,-1,-1,-1,-1,-1,-1,-1,-1,-1,-1,-1,-1,-1,-1,-1},
  {4,5,6,7,8,4,5,6,7,8,4,5,6,7,8,0,-1,-1,-1,-1,-1,-1,-1,-1,-1,-1,-1,-1},
  {1,2,3,1,2,3,1,2,3,1,2,3,1,2,3,-1,-1,-1,-1,-1,-1,-1,-1,-1,-1,-1,-1,-1},
  {4,5,6,7,8,4,5,6,7,8,4,5,6,7,8,4,5,6,7,8,4,5,6,7,8,0,0,0},
  {4,5,6,7,8,0,0,0,0,0,-1,-1,-1,-1,-1,-1,-1,-1,-1,-1,-1,-1,-1,-1,-1,-1,-1,-1}
};

// 19 output path segments: stage column of M=0, l3, path index within l3 group
constexpr int SEG_L3[19]  = {0,1,2,1,1,2,3,0,1,2,3,0,1,2,3,4,1,2,2};
constexpr int SEG_COL[19] = {0,1,4,9,12,15,20,27,28,31,36,43,44,47,52,59,68,71,76};
constexpr int SEG_P[19]   = {1,2,1,0,3,2,0,0,4,4,1,2,5,5,2,0,1,0,3};
constexpr int L3BASE_C[5] = {0, 3145728, 22020096, 53477376, 75497472};
constexpr int PCNT_C[5]   = {3, 6, 6, 3, 1};

constexpr int STAGE_W   = 84;          // 81 cols padded (bank spread)
constexpr int WAVE_LDS  = 16 * STAGE_W;

template<int SG, int K>
DEVFN float zval(const float (&x1c)[9], const float (&x2c)[9]) {
  constexpr int i1 = ZI1[SG][K];
  if constexpr (i1 < 0) { return 0.0f; }
  else {
    constexpr int i2 = ZI2[SG][K];
    return x1c[i1] * x2c[i2];
  }
}

// ---------------- per-super-GEMM compute, C -> LDS stage --------------------
template<int SG>
DEVFN void run_sg(const float (&x1c)[9], const float (&x2c)[9],
                  const float* __restrict__ Bws, float* lds_wave,
                  int lane, int hf)
{
  constexpr int KC = SG_KC[SG];
  constexpr int NT = SG_NT[SG];
  const float* __restrict__ bsg = Bws + SG_BOFF[SG];

  v8f acc[NT] = {};

  sfor<0, KC>([&](auto kcT) {
    constexpr int kc = decltype(kcT)::v;
    // A chunk: low half-wave gets k = 4kc,4kc+1 ; high half gets 4kc+2,4kc+3
    const float zl0 = zval<SG, 4*kc + 0>(x1c, x2c);
    const float zl1 = zval<SG, 4*kc + 1>(x1c, x2c);
    const float zh0 = zval<SG, 4*kc + 2>(x1c, x2c);
    const float zh1 = zval<SG, 4*kc + 3>(x1c, x2c);
    v2f a;
    a.x = hf ? zh0 : zl0;
    a.y = hf ? zh1 : zl1;
    sfor<0, NT>([&](auto ntT) {
      constexpr int nt = decltype(ntT)::v;
      const int off = (kc * NT + nt) * 64 + (lane << 1);
      v2f bv;
      bv.x = bsg[off];
      bv.y = bsg[off + 1];
      acc[nt] = __builtin_amdgcn_wmma_f32_16x16x4_f32(
          false, a, false, bv, (short)0, acc[nt], false, false);
    });
  });

  // Stage C: vgpr j holds (row j + 8*hf, col nt*16 + (lane&15)).
  sfor<0, NT>([&](auto ntT) {
    constexpr int nt = decltype(ntT)::v;
    const int c = nt * 16 + (lane & 15);
    if (c < SG_N[SG]) {                       // mask padded columns
      float* p = lds_wave + (hf * 8) * STAGE_W + (SG_CBASE[SG] + c);
      sfor<0, 8>([&](auto jT) {
        constexpr int j = decltype(jT)::v;
        p[j * STAGE_W] = acc[nt][j];
      });
    }
  });
}

// ---------------- main kernel: one wave per 16-row tile ---------------------
__global__ __launch_bounds__(256) void tp_main(
    const float* __restrict__ x10, const float* __restrict__ x11,
    const float* __restrict__ x12, const float* __restrict__ x20,
    const float* __restrict__ x21, const float* __restrict__ x22,
    const float* __restrict__ Bws, float* __restrict__ out)
{
  __shared__ float stage[8 * WAVE_LDS];     // 43008 B: 8 waves x 16 x 84

  const int lane = threadIdx.x & 31;
  const int tile = (blockIdx.x * 256 + threadIdx.x) >> 5;   // 65536 tiles
  const int hf   = lane >> 4;
  const int b    = tile >> 2;            // row = b*64 + i ; 4 tiles per b
  const int i0   = (tile & 3) << 4;
  const int row  = (tile << 4) + (lane & 15);   // lanes L and L+16 share a row

  float* lds_wave = stage + (threadIdx.x >> 5) * WAVE_LDS;

  float x1c[9], x2c[9];
  x1c[0] = x10[row];
  x2c[0] = x20[row];
  sfor<0, 3>([&](auto mT) { constexpr int m = decltype(mT)::v;
    x1c[1 + m] = x11[row * 3 + m]; x2c[1 + m] = x21[row * 3 + m]; });
  sfor<0, 5>([&](auto mT) { constexpr int m = decltype(mT)::v;
    x1c[4 + m] = x12[row * 5 + m]; x2c[4 + m] = x22[row * 5 + m]; });

  run_sg<0>(x1c, x2c, Bws, lds_wave, lane, hf);
  run_sg<1>(x1c, x2c, Bws, lds_wave, lane, hf);
  run_sg<2>(x1c, x2c, Bws, lds_wave, lane, hf);
  run_sg<3>(x1c, x2c, Bws, lds_wave, lane, hf);
  run_sg<4>(x1c, x2c, Bws, lds_wave, lane, hf);

  // Same-wave LDS writes -> reads are in order (DScnt); no barrier needed.
  // Each path segment is contiguous 16*(2l3+1) floats in global memory.
  sfor<0, 19>([&](auto sT) {
    constexpr int s   = decltype(sT)::v;
    constexpr int l3  = SEG_L3[s];
    constexpr int w3  = 2 * l3 + 1;
    constexpr int len = 16 * w3;
    constexpr int col = SEG_COL[s];
    constexpr int p   = SEG_P[s];
    const int gbase = L3BASE_C[l3] + (b * (PCNT_C[l3] * 64) + p * 64 + i0) * w3;
    sfor<0, (len + 31) / 32>([&](auto itT) {
      constexpr int it = decltype(itT)::v;
      const int q = lane + it * 32;              // flat (i_local, M) index
      if ((it + 1) * 32 <= len || q < len) {
        const int il = q / w3;                   // w3 is compile-time
        const int M  = q - il * w3;
        out[gbase + q] = lds_wave[il * STAGE_W + col + M];
      }
    });
  });
}

// ---------------- setup kernel: build Bws (w3j in WMMA B lane layout) -------
struct W3JP { const float* p[19]; };

__device__ const int d_SG_BEND[5]    = {192, 448, 704, 1600, 1792};
__device__ const int d_SG_BOFF[5]    = {0, 192, 448, 704, 1600};
__device__ const int d_SG_NT[5]      = {1, 1, 1, 2, 1};
__device__ const int d_SG_NB[5]      = {2, 2, 1, 2, 2};
__device__ const int d_SGB_OFF[5][2] = {{0,9},{0,15},{0,0},{0,25},{0,5}};
__device__ const int d_SGB_L1[5][2]  = {{1,0},{1,0},{2,0},{2,1},{0,2}};
__device__ const int d_SGB_L2[5][2]  = {{1,1},{2,0},{1,0},{2,0},{2,0}};
// w3j input index (d_in[6 + idx]) per (l1,l2,l3), -1 = no triple
__device__ const int d_TRIP[3][3][5] = {
  { { 0,-1,-1,-1,-1}, {-1, 1,-1,-1,-1}, {-1,-1, 2,-1,-1} },
  { {-1, 3,-1,-1,-1}, { 4, 5, 6,-1,-1}, {-1, 7, 8, 9,-1} },
  { {-1,-1,10,-1,-1}, {-1,11,12,13,-1}, {14,15,16,17,18} } };

__global__ void tp_setup(W3JP w, float* __restrict__ Bws)
{
  const int tid = blockIdx.x * blockDim.x + threadIdx.x;
  const int nth = gridDim.x * blockDim.x;

  // B matrix values in per-lane WMMA layout: entry = (kc*NT+nt)*64 + lane*2 + j
  for (int e = tid; e < 1792; e += nth) {
    int sg = 0;
    while (sg < 4 && e >= d_SG_BEND[sg]) ++sg;
    const int f     = e - d_SG_BOFF[sg];
    const int NTg   = d_SG_NT[sg];
    const int chunk = f >> 6;
    const int kc    = chunk / NTg;
    const int nt    = chunk - kc * NTg;
    const int rem   = f & 63;
    const int lane  = rem >> 1;
    const int j     = rem & 1;
    const int k     = kc * 4 + ((lane >> 4) << 1) + j;
    const int c     = nt * 16 + (lane & 15);
    float val = 0.0f;
    for (int bi = 0; bi < d_SG_NB[sg]; ++bi) {
      const int off = d_SGB_OFF[sg][bi];
      const int l1  = d_SGB_L1[sg][bi];
      const int l2  = d_SGB_L2[sg][bi];
      const int sz  = (2 * l1 + 1) * (2 * l2 + 1);   // square blocks
      if (k >= off && k < off + sz && c >= off && c < off + sz) {
        const int kl = k - off;
        int cl = c - off;
        const int l3min = l1 > l2 ? l1 - l2 : l2 - l1;
        for (int l3 = l3min; l3 <= l1 + l2; ++l3) {
          const int w3 = 2 * l3 + 1;
          if (cl < w3) { val = w.p[d_TRIP[l1][l2][l3]][kl * w3 + cl]; break; }
          cl -= w3;
        }
      }
    }
    Bws[e] = val;
  }
}

// ---------------- launch ----------------------------------------------------
extern "C" void kernel_launch(void* const* d_in, const int* in_sizes, int n_in,
                              void* d_out, int out_size, void* d_ws, size_t ws_size,
                              hipStream_t stream)
{
  (void)in_sizes; (void)n_in; (void)out_size; (void)ws_size;
  const float* x10 = (const float*)d_in[0];
  const float* x11 = (const float*)d_in[1];
  const float* x12 = (const float*)d_in[2];
  const float* x20 = (const float*)d_in[3];
  const float* x21 = (const float*)d_in[4];
  const float* x22 = (const float*)d_in[5];
  W3JP w;
  for (int t = 0; t < 19; ++t) w.p[t] = (const float*)d_in[6 + t];

  float* Bws = (float*)d_ws;              // 1792 floats of w3j in B lane layout

  tp_setup<<<8, 256, 0, stream>>>(w, Bws);

  const int tiles  = (16384 * 64) / 16;   // 65536 row-tiles, 1 wave each
  const int blocks = tiles / 8;           // 8 waves per 256-thread block
  tp_main<<<blocks, 256, 0, stream>>>(x10, x11, x12, x20, x21, x22,
                                      Bws, (float*)d_out);
}